// GCN_64690797412512
// MI455X (gfx1250) — compile-verified
//
#include <hip/hip_runtime.h>
#include <stdint.h>

typedef __attribute__((ext_vector_type(2))) float v2f;
typedef __attribute__((ext_vector_type(8))) float v8f;

#define NUM_NODES  102400
#define NUM_EDGES  3276800
#define NUM_GRAPHS 1024

// ---------------------------------------------------------------------------
// Utility: zero a float buffer (grid-stride)
// ---------------------------------------------------------------------------
__global__ void zero_f32(float* __restrict__ p, int n) {
    int i = blockIdx.x * blockDim.x + threadIdx.x;
    int stride = gridDim.x * blockDim.x;
    for (; i < n; i += stride) p[i] = 0.0f;
}

// ---------------------------------------------------------------------------
// deg[src] += edge_attr  (one thread per edge)
// ---------------------------------------------------------------------------
__global__ void deg_kernel(const long long* __restrict__ ei,
                           const float* __restrict__ attr,
                           float* __restrict__ deg) {
    int e = blockIdx.x * blockDim.x + threadIdx.x;
    if (e >= NUM_EDGES) return;
    int s = (int)ei[e];
    atomicAdd(&deg[s], attr[e]);
}

// deg -> dinv in place
__global__ void dinv_kernel(float* __restrict__ deg) {
    int i = blockIdx.x * blockDim.x + threadIdx.x;
    if (i >= NUM_NODES) return;
    float d = deg[i];
    deg[i] = (d > 0.0f) ? rsqrtf(d) : 0.0f;
}

// norm_w[e] = -dinv[src] * attr[e] * dinv[dst]   (lambda_max = 2 -> diag = 0)
__global__ void edgew_kernel(const long long* __restrict__ ei,
                             const float* __restrict__ attr,
                             const float* __restrict__ dinv,
                             float* __restrict__ nw) {
    int e = blockIdx.x * blockDim.x + threadIdx.x;
    if (e >= NUM_EDGES) return;
    int s = (int)ei[e];
    int d = (int)ei[NUM_EDGES + e];
    nw[e] = -dinv[s] * attr[e] * dinv[d];
}

// ---------------------------------------------------------------------------
// SpMV scatter: Tout[dst] += nw[e] * Tin[src]   (one wave32 per edge)
// Coalesced 128B gather from Tin row, coalesced f32 atomics into Tout row.
// ---------------------------------------------------------------------------
template <int F>
__global__ void spmv_kernel(const long long* __restrict__ ei,
                            const float* __restrict__ nw,
                            const float* __restrict__ tin,
                            float* __restrict__ tout) {
    int gid  = blockIdx.x * blockDim.x + threadIdx.x;
    int e    = gid >> 5;            // one edge per wave (wave32)
    int lane = threadIdx.x & 31;
    if (e >= NUM_EDGES) return;
    int s = (int)ei[e];
    int d = (int)ei[NUM_EDGES + e];
    float w = nw[e];
    const float* __restrict__ xin = tin + (long)s * F;
    float* __restrict__ xo = tout + (long)d * F;
#pragma unroll
    for (int f = lane; f < F; f += 32)
        atomicAdd(&xo[f], w * xin[f]);
}

// ---------------------------------------------------------------------------
// Fused Chebyshev GEMM + bias + ReLU using V_WMMA_F32_16X16X4_F32:
//   OUT = relu(T0 @ W[0] + T1 @ W[1] + (2*S - T0) @ W[2] + bias)
// One wave computes one 16(M)x16(N) fp32 output tile; K-loop steps of 4.
// A layout (16x4 f32): lanes 0-15 hold M=0..15, v0=K0/v1=K1; lanes 16-31: K2/K3.
// B layout (4x16 f32): row K in VGPR K%2, lane half selected by K/2.
// C/D (16x16 f32): VGPR j -> M=j (lanes 0-15) / M=j+8 (lanes 16-31), N=lane&15.
// ---------------------------------------------------------------------------
template <int FIN, int FOUT>
__global__ void cheb_gemm_relu(const float* __restrict__ T0,
                               const float* __restrict__ T1,
                               const float* __restrict__ S,
                               const float* __restrict__ W,   // [3, FIN, FOUT]
                               const float* __restrict__ bias,
                               float* __restrict__ OUT) {
    constexpr int NT = FOUT / 16;
    int wid  = (blockIdx.x * blockDim.x + threadIdx.x) >> 5;
    int lane = threadIdx.x & 31;
    int tileM = wid / NT;
    int tileN = (wid % NT) * 16;
    if (tileM >= NUM_NODES / 16) return;
    int row16 = tileM * 16;

    int r    = lane & 15;          // M row (A) / N col (B,C)
    int kofs = (lane >> 4) << 1;   // K offset: 0 for lanes 0-15, 2 for 16-31

    const float* __restrict__ A0 = T0 + (long)(row16 + r) * FIN;
    const float* __restrict__ A1 = T1 + (long)(row16 + r) * FIN;
    const float* __restrict__ A2 = S  + (long)(row16 + r) * FIN;

    v8f acc = {};

    // term 0: T0 @ W0
    {
        const float* __restrict__ Wt = W;
        for (int k0 = 0; k0 < FIN; k0 += 4) {
            int ka = k0 + kofs;
            v2f a, b;
            a.x = A0[ka];
            a.y = A0[ka + 1];
            b.x = Wt[(long)ka * FOUT + tileN + r];
            b.y = Wt[(long)(ka + 1) * FOUT + tileN + r];
            acc = __builtin_amdgcn_wmma_f32_16x16x4_f32(
                false, a, false, b, (short)0, acc, false, false);
        }
    }
    // term 1: T1 @ W1
    {
        const float* __restrict__ Wt = W + (long)FIN * FOUT;
        for (int k0 = 0; k0 < FIN; k0 += 4) {
            int ka = k0 + kofs;
            v2f a, b;
            a.x = A1[ka];
            a.y = A1[ka + 1];
            b.x = Wt[(long)ka * FOUT + tileN + r];
            b.y = Wt[(long)(ka + 1) * FOUT + tileN + r];
            acc = __builtin_amdgcn_wmma_f32_16x16x4_f32(
                false, a, false, b, (short)0, acc, false, false);
        }
    }
    // term 2: (2*S - T0) @ W2   (fused Chebyshev recurrence, diag == 0)
    {
        const float* __restrict__ Wt = W + 2L * FIN * FOUT;
        for (int k0 = 0; k0 < FIN; k0 += 4) {
            int ka = k0 + kofs;
            v2f a, b;
            a.x = 2.0f * A2[ka]     - A0[ka];
            a.y = 2.0f * A2[ka + 1] - A0[ka + 1];
            b.x = Wt[(long)ka * FOUT + tileN + r];
            b.y = Wt[(long)(ka + 1) * FOUT + tileN + r];
            acc = __builtin_amdgcn_wmma_f32_16x16x4_f32(
                false, a, false, b, (short)0, acc, false, false);
        }
    }

    int mhi = (lane >> 4) << 3;    // +8 rows for lanes 16-31
    int cc  = tileN + r;
    float bv = bias[cc];
#pragma unroll
    for (int j = 0; j < 8; ++j) {
        int rr = row16 + j + mhi;
        float v = acc[j] + bv;
        OUT[(long)rr * FOUT + cc] = v > 0.0f ? v : 0.0f;
    }
}

// ---------------------------------------------------------------------------
// Global mean pool: scatter sums + counts (one wave32 per node)
// ---------------------------------------------------------------------------
__global__ void pool_scatter(const float* __restrict__ H,
                             const long long* __restrict__ batch,
                             float* __restrict__ sums,
                             float* __restrict__ cnts) {
    int gid  = blockIdx.x * blockDim.x + threadIdx.x;
    int node = gid >> 5;
    int lane = threadIdx.x & 31;
    if (node >= NUM_NODES) return;
    int g = (int)batch[node];
    const float* __restrict__ h = H + (long)node * 128;
    float* __restrict__ s = sums + (long)g * 128;
#pragma unroll
    for (int f = lane; f < 128; f += 32)
        atomicAdd(&s[f], h[f]);
    if (lane == 0) atomicAdd(&cnts[g], 1.0f);
}

// Per-graph: out[g] = (sums[g]/max(cnt,1)) @ lin_w + lin_b   (one block/graph)
__global__ void pool_final(const float* __restrict__ sums,
                           const float* __restrict__ cnts,
                           const float* __restrict__ lin_w,  // [128,2]
                           const float* __restrict__ lin_b,  // [2]
                           float* __restrict__ out) {
    __shared__ float red0[128];
    __shared__ float red1[128];
    int g = blockIdx.x;
    int t = threadIdx.x;   // 128 threads
    float c = cnts[g];
    c = c > 1.0f ? c : 1.0f;
    float v = sums[(long)g * 128 + t] / c;
    red0[t] = v * lin_w[t * 2 + 0];
    red1[t] = v * lin_w[t * 2 + 1];
    __syncthreads();
    for (int s = 64; s > 0; s >>= 1) {
        if (t < s) { red0[t] += red0[t + s]; red1[t] += red1[t + s]; }
        __syncthreads();
    }
    if (t == 0) {
        out[g * 2 + 0] = red0[0] + lin_b[0];
        out[g * 2 + 1] = red1[0] + lin_b[1];
    }
}

// ---------------------------------------------------------------------------
// Host launch
// ---------------------------------------------------------------------------
extern "C" void kernel_launch(void* const* d_in, const int* in_sizes, int n_in,
                              void* d_out, int out_size, void* d_ws, size_t ws_size,
                              hipStream_t stream) {
    const float*     x     = (const float*)d_in[0];
    const long long* ei    = (const long long*)d_in[1];
    const float*     attr  = (const float*)d_in[2];
    const long long* batch = (const long long*)d_in[3];
    const float*     w1    = (const float*)d_in[4];   // [3,128,64]
    const float*     b1    = (const float*)d_in[5];
    const float*     w2    = (const float*)d_in[6];   // [3,64,64]
    const float*     b2    = (const float*)d_in[7];
    const float*     w3    = (const float*)d_in[8];   // [3,64,128]
    const float*     b3    = (const float*)d_in[9];
    const float*     lw    = (const float*)d_in[10];  // [128,2]
    const float*     lb    = (const float*)d_in[11];
    float*           out   = (float*)d_out;

    // Workspace carve-out (fp32 elements)
    char* ws = (char*)d_ws;
    size_t off = 0;
    auto carve = [&](size_t bytes) {
        float* p = (float*)(ws + off);
        off += (bytes + 255) & ~(size_t)255;
        return p;
    };
    float* deg  = carve((size_t)NUM_NODES * 4);            // then dinv in place
    float* nw   = carve((size_t)NUM_EDGES * 4);
    float* T1   = carve((size_t)NUM_NODES * 128 * 4);
    float* S    = carve((size_t)NUM_NODES * 128 * 4);
    float* Ha   = carve((size_t)NUM_NODES * 128 * 4);
    float* Hb   = carve((size_t)NUM_NODES * 128 * 4);
    float* psum = carve((size_t)NUM_GRAPHS * 128 * 4);
    float* pcnt = carve((size_t)NUM_GRAPHS * 4);
    (void)ws_size; (void)in_sizes; (void)n_in; (void)out_size;

    const int TB = 256;
    const int edgeBlocks  = (NUM_EDGES + TB - 1) / TB;           // 1 thread / edge
    const int edgeWaveBlk = (NUM_EDGES * 32) / TB;               // 1 wave / edge
    const int nodeBlocks  = (NUM_NODES + TB - 1) / TB;
    const int nodeWaveBlk = (NUM_NODES * 32) / TB;
    const int ZB = 2048;                                          // zero-kernel blocks

    // --- normalization ---
    zero_f32<<<ZB, TB, 0, stream>>>(deg, NUM_NODES);
    deg_kernel<<<edgeBlocks, TB, 0, stream>>>(ei, attr, deg);
    dinv_kernel<<<nodeBlocks, TB, 0, stream>>>(deg);
    edgew_kernel<<<edgeBlocks, TB, 0, stream>>>(ei, attr, deg, nw);

    // --- layer 1: Fin=128 -> Fout=64, T0 = x, OUT = Ha ---
    zero_f32<<<ZB, TB, 0, stream>>>(T1, NUM_NODES * 128);
    spmv_kernel<128><<<edgeWaveBlk, TB, 0, stream>>>(ei, nw, x, T1);
    zero_f32<<<ZB, TB, 0, stream>>>(S, NUM_NODES * 128);
    spmv_kernel<128><<<edgeWaveBlk, TB, 0, stream>>>(ei, nw, T1, S);
    cheb_gemm_relu<128, 64><<<(NUM_NODES / 16) * (64 / 16) / 8, TB, 0, stream>>>(
        x, T1, S, w1, b1, Ha);

    // --- layer 2: Fin=64 -> Fout=64, T0 = Ha, OUT = Hb ---
    zero_f32<<<ZB, TB, 0, stream>>>(T1, NUM_NODES * 64);
    spmv_kernel<64><<<edgeWaveBlk, TB, 0, stream>>>(ei, nw, Ha, T1);
    zero_f32<<<ZB, TB, 0, stream>>>(S, NUM_NODES * 64);
    spmv_kernel<64><<<edgeWaveBlk, TB, 0, stream>>>(ei, nw, T1, S);
    cheb_gemm_relu<64, 64><<<(NUM_NODES / 16) * (64 / 16) / 8, TB, 0, stream>>>(
        Ha, T1, S, w2, b2, Hb);

    // --- layer 3: Fin=64 -> Fout=128, T0 = Hb, OUT = Ha ---
    zero_f32<<<ZB, TB, 0, stream>>>(T1, NUM_NODES * 64);
    spmv_kernel<64><<<edgeWaveBlk, TB, 0, stream>>>(ei, nw, Hb, T1);
    zero_f32<<<ZB, TB, 0, stream>>>(S, NUM_NODES * 64);
    spmv_kernel<64><<<edgeWaveBlk, TB, 0, stream>>>(ei, nw, T1, S);
    cheb_gemm_relu<64, 128><<<(NUM_NODES / 16) * (128 / 16) / 8, TB, 0, stream>>>(
        Hb, T1, S, w3, b3, Ha);

    // --- global mean pool + linear head ---
    zero_f32<<<ZB, TB, 0, stream>>>(psum, NUM_GRAPHS * 128 + NUM_GRAPHS);
    pool_scatter<<<nodeWaveBlk, TB, 0, stream>>>(Ha, batch, psum, pcnt);
    pool_final<<<NUM_GRAPHS, 128, 0, stream>>>(psum, pcnt, lw, lb, out);
}